// Model_69853348102672
// MI455X (gfx1250) — compile-verified
//
#include <hip/hip_runtime.h>
#include <stdint.h>

// ---------------------------------------------------------------------------
// Affine scan  h_t = a_t*h_{t-1} + b_t  over L, per (B,D) channel.
// B=8, L=8192, D=256, f32. Memory-bound (0.17 FLOP/byte) -> optimize the
// data-movement path: TDM async tiles into LDS, double buffered, 512 blocks.
// Phase 1: per-chunk affine summaries (reads a,b -> warms L2, 134 MB < 192 MB)
// Phase 2: tiny scan over chunk summaries -> exclusive carries
// Phase 3: apply carries, re-read a,b (L2 hits), NT-store h (no L2 pollution)
// ---------------------------------------------------------------------------

typedef unsigned int u32x4 __attribute__((ext_vector_type(4)));
typedef int          i32x4 __attribute__((ext_vector_type(4)));
typedef int          i32x8 __attribute__((ext_vector_type(8)));

constexpr int kL      = 8192;
constexpr int kD      = 256;
constexpr int kNChunk = 64;            // chunks per sequence
constexpr int kCLen   = kL / kNChunk;  // 128 timesteps per chunk
constexpr int kTS     = 8;             // timesteps per LDS tile
constexpr int kNT     = kCLen / kTS;   // 16 tiles per chunk
constexpr int kTileElems = kTS * kD;   // 2048 floats = 8 KB, contiguous in memory

#if __has_builtin(__builtin_amdgcn_tensor_load_to_lds)
#define USE_TDM 1
#else
#define USE_TDM 0
#endif

#if USE_TDM
// Low 32 bits of a generic pointer to __shared__ == LDS byte offset.
__device__ __forceinline__ uint32_t lds_byte_off(const void* p) {
  return (uint32_t)(uintptr_t)p;
}

// Issue one TDM transfer: 2048 contiguous f32 (one 1-row tile) global -> LDS.
// Descriptor layout per CDNA5 ISA ch.8 (D# groups 0/1; groups 2/3 unused->0).
__device__ __forceinline__ void tdm_load_tile(const float* gsrc, uint32_t lds_addr) {
  const uint64_t ga = (uint64_t)(uintptr_t)gsrc;
  u32x4 g0;
  g0.x = 1u;                                   // count=1 (valid), user mode, no gather
  g0.y = lds_addr;                             // [63:32]  lds_addr (bytes)
  g0.z = (uint32_t)ga;                         // [95:64]  global_addr lo
  g0.w = (uint32_t)((ga >> 32) & 0x01FFFFFFu)  // [120:96] global_addr hi
       | (2u << 30);                           // [127:126] type=2 ("image")
  i32x8 g1;
  g1[0] = (int)(2u << 16);                     // data_size=2 (4 bytes); mask/pad/iterate=0
  g1[1] = (int)((kTileElems & 0xFFFF) << 16);  // tensor_dim0[15:0] in [63:48]
  g1[2] = (int)((kTileElems >> 16) | (1u << 16)); // tensor_dim0 hi | tensor_dim1=1
  g1[3] = (int)((kTileElems & 0xFFFF) << 16);  // tile_dim0 = 2048
  g1[4] = 1;                                   // tile_dim1=1, tile_dim2=0 (unused)
  g1[5] = kTileElems;                          // tensor_dim0_stride lo (single row)
  g1[6] = 0;                                   // stride0 hi | stride1 lo
  g1[7] = 0;                                   // stride1 hi
  const i32x4 gz = {0, 0, 0, 0};
#if defined(__clang_major__) && (__clang_major__ >= 23)
  const i32x8 gz8 = {0, 0, 0, 0, 0, 0, 0, 0};
  __builtin_amdgcn_tensor_load_to_lds(g0, g1, gz, gz, gz8, 0);
#else
  __builtin_amdgcn_tensor_load_to_lds(g0, g1, gz, gz, 0);
#endif
}
#endif  // USE_TDM

// Shared body. PHASE1=true: per-chunk summary (A_c = prod a, B_c = scan tail).
// PHASE1=false: apply scan within chunk from the exclusive carry, NT-store h.
template <bool PHASE1>
__device__ __forceinline__
void scan_chunk_body(const float* __restrict__ ga_base,
                     const float* __restrict__ gb_base,
                     float* __restrict__ out,
                     float* __restrict__ wsA,
                     float* __restrict__ wsB) {
  __shared__ float lds_a[2][kTileElems];   // 2 x 8 KB
  __shared__ float lds_b[2][kTileElems];   // 2 x 8 KB  (32 KB total)

  const int c    = blockIdx.x;             // chunk index
  const int bidx = blockIdx.y;             // batch index
  const int d    = threadIdx.x;            // channel (0..255)
  const size_t base = ((size_t)bidx * kL + (size_t)c * kCLen) * (size_t)kD;
  const float* ga = ga_base + base;        // chunk is kCLen*kD contiguous floats
  const float* gb = gb_base + base;

#if USE_TDM
  // TDM ignores EXEC: must use a true scalar (SALU) branch so only wave 0
  // issues the DMA -- an EXEC-masked vector branch would issue 8 duplicates.
  const bool is_wave0 = ((__builtin_amdgcn_readfirstlane((int)threadIdx.x) >> 5) == 0);
  if (is_wave0) {
    tdm_load_tile(ga, lds_byte_off(&lds_a[0][0]));
    tdm_load_tile(gb, lds_byte_off(&lds_b[0][0]));
  }
#endif

  float h = 0.0f;
  if (!PHASE1) {
    // exclusive carry written by chunk_scan_kernel into wsB
    h = wsB[((size_t)bidx * kNChunk + c) * kD + d];
  }
  float Aacc = 1.0f;

  int buf = 0;
  for (int i = 0; i < kNT; ++i) {
#if USE_TDM
    if (is_wave0) {
      if (i + 1 < kNT) {
        // prefetch next tile into the other buffer (its last reader passed
        // the trailing barrier of iteration i-1)
        tdm_load_tile(ga + (size_t)(i + 1) * kTileElems,
                      lds_byte_off(&lds_a[buf ^ 1][0]));
        tdm_load_tile(gb + (size_t)(i + 1) * kTileElems,
                      lds_byte_off(&lds_b[buf ^ 1][0]));
        __builtin_amdgcn_s_wait_tensorcnt(2);  // in-order: oldest pair (tile i) done
      } else {
        __builtin_amdgcn_s_wait_tensorcnt(0);  // drain
      }
    }
    __syncthreads();                           // publish LDS tile i to all waves
#else
    for (int j = threadIdx.x; j < kTileElems; j += kD) {
      lds_a[buf][j] = ga[(size_t)i * kTileElems + j];
      lds_b[buf][j] = gb[(size_t)i * kTileElems + j];
    }
    __syncthreads();
#endif

#pragma unroll
    for (int t = 0; t < kTS; ++t) {
      const float av = lds_a[buf][t * kD + d];
      const float bv = lds_b[buf][t * kD + d];
      h = fmaf(av, h, bv);                     // h = a*h + b
      if (PHASE1) {
        Aacc *= av;
      } else {
        // NT store: don't let the 67 MB output stream evict the L2-resident
        // a/b working set (134 MB) that phase 3 is re-reading.
        __builtin_nontemporal_store(h, &out[base + (size_t)(i * kTS + t) * kD + d]);
      }
    }
    __syncthreads();                           // tile i fully consumed
    buf ^= 1;
  }

  if (PHASE1) {
    const size_t sidx = ((size_t)bidx * kNChunk + c) * kD + d;
    wsA[sidx] = Aacc;   // A_c = prod of a over chunk
    wsB[sidx] = h;      // B_c = local scan value at chunk end
  }
}

// Defined first so the TDM path shows at the top of the device assembly.
__global__ __launch_bounds__(kD)
void scan_phase1_kernel(const float* __restrict__ a, const float* __restrict__ b,
                        float* __restrict__ out,
                        float* __restrict__ wsA, float* __restrict__ wsB) {
  scan_chunk_body<true>(a, b, out, wsA, wsB);
}

__global__ __launch_bounds__(kD)
void scan_phase3_kernel(const float* __restrict__ a, const float* __restrict__ b,
                        float* __restrict__ out,
                        float* __restrict__ wsA, float* __restrict__ wsB) {
  scan_chunk_body<false>(a, b, out, wsA, wsB);
}

// PHASE 2: sequential scan over the 64 chunk summaries per (b,d) channel.
// Overwrites wsB[c] with the EXCLUSIVE carry entering chunk c.
// Tiny: 8 blocks x 256 threads x 64 steps, fully coalesced per step.
__global__ __launch_bounds__(kD)
void chunk_scan_kernel(const float* __restrict__ wsA, float* __restrict__ wsB) {
  const int bidx = blockIdx.x;
  const int d    = threadIdx.x;
  float run = 0.0f;                            // h before chunk 0 is 0 (h_1 = b_1)
  for (int c = 0; c < kNChunk; ++c) {
    const size_t idx = ((size_t)bidx * kNChunk + c) * kD + d;
    const float Ac = wsA[idx];
    const float Bc = wsB[idx];
    wsB[idx] = run;                            // exclusive carry for chunk c
    run = fmaf(Ac, run, Bc);                   // inclusive: B_full_c = A_c*run + B_c
  }
}

extern "C" void kernel_launch(void* const* d_in, const int* in_sizes, int n_in,
                              void* d_out, int out_size, void* d_ws, size_t ws_size,
                              hipStream_t stream) {
  (void)n_in; (void)out_size; (void)ws_size;
  const float* a = (const float*)d_in[0];
  const float* b = (const float*)d_in[1];
  float* out = (float*)d_out;

  const int Btot = in_sizes[0] / (kL * kD);    // = 8 for the reference shapes

  // workspace: two f32 arrays of [B, NCHUNK, D]  (1 MB total for B=8)
  float* wsA = (float*)d_ws;
  float* wsB = wsA + (size_t)Btot * kNChunk * kD;

  dim3 grid(kNChunk, Btot);                    // 512 blocks x 256 threads (8 waves)
  scan_phase1_kernel<<<grid, kD, 0, stream>>>(a, b, out, wsA, wsB);
  chunk_scan_kernel<<<Btot, kD, 0, stream>>>(wsA, wsB);
  scan_phase3_kernel<<<grid, kD, 0, stream>>>(a, b, out, wsA, wsB);
}